// GAT_52183852646441
// MI455X (gfx1250) — compile-verified
//
#include <hip/hip_runtime.h>
#include <hip/hip_bf16.h>
#include <math.h>

typedef __attribute__((ext_vector_type(2))) float v2f;
typedef __attribute__((ext_vector_type(8))) float v8f;

#define NEG_SLOPE 0.2f
#define BN_EPS 1e-5f

__device__ __forceinline__ float lrelu(float x) { return x > 0.f ? x : NEG_SLOPE * x; }

// float atomic max via monotonic int mapping (init value must be -inf)
__device__ __forceinline__ void atomicMaxF(float* addr, float val) {
  if (val >= 0.f) atomicMax((int*)addr, __float_as_int(val));
  else            atomicMin((unsigned int*)addr, __float_as_uint(val));
}

// ---------------- GEMM1: h1[N,64] = x[N,128] @ W1[128,64] via V_WMMA_F32_16X16X4_F32
// LDS holds W1 repacked as K-pairs: sW[p*160 + col*2 + j] = W1[(2p+j)*64 + col]
// (row stride 160 floats -> 32-bank shift per pair-row: half-waves hit disjoint banks)
__global__ void __launch_bounds__(256) gemm1_kernel(const float* __restrict__ x,
                                                    const float* __restrict__ W1,
                                                    float* __restrict__ h1, int n) {
  __shared__ float sW[64 * 160];  // 40 KB
  for (int i = threadIdx.x; i < 64 * 128; i += 256) {
    int p = i >> 7, r = i & 127;
    sW[p * 160 + r] = W1[(2 * p + (r & 1)) * 64 + (r >> 1)];
  }
  __syncthreads();
  const int wave = threadIdx.x >> 5;
  const int lane = threadIdx.x & 31;
  const int row0 = (blockIdx.x * 8 + wave) * 16;
  if (row0 >= n) return;                      // wave-uniform; EXEC all-ones inside
  const int koff = (lane >= 16) ? 2 : 0;      // A: lanes16-31 hold K=2,3
  const int l15 = lane & 15;
  const float* arow = x + (size_t)(row0 + l15) * 128 + koff;
  const float* brow = sW + (koff >> 1) * 160 + l15 * 2;  // pair index = k/2 + (lane>=16)
  v8f acc0 = {}, acc1 = {}, acc2 = {}, acc3 = {};
#pragma unroll 4
  for (int k = 0; k < 128; k += 4) {
    v2f a = *(const v2f*)(arow + k);
    const float* bp = brow + (k >> 1) * 160;
    v2f b0 = *(const v2f*)(bp + 0);
    v2f b1 = *(const v2f*)(bp + 32);
    v2f b2 = *(const v2f*)(bp + 64);
    v2f b3 = *(const v2f*)(bp + 96);
    acc0 = __builtin_amdgcn_wmma_f32_16x16x4_f32(false, a, false, b0, (short)0, acc0, false, false);
    acc1 = __builtin_amdgcn_wmma_f32_16x16x4_f32(false, a, false, b1, (short)0, acc1, false, false);
    acc2 = __builtin_amdgcn_wmma_f32_16x16x4_f32(false, a, false, b2, (short)0, acc2, false, false);
    acc3 = __builtin_amdgcn_wmma_f32_16x16x4_f32(false, a, false, b3, (short)0, acc3, false, false);
  }
  float* outb = h1 + (size_t)row0 * 64;
  const int mb = (lane >> 4) * 8;  // C/D: VGPR i = M=i (lanes 0-15) / M=8+i (lanes 16-31)
#pragma unroll
  for (int m = 0; m < 8; ++m) {
    float* orow = outb + (size_t)(mb + m) * 64 + l15;
    orow[0] = acc0[m]; orow[16] = acc1[m]; orow[32] = acc2[m]; orow[48] = acc3[m];
  }
}

// ---------------- GEMM2: h2[N,40] = hbn[N,64] @ W2[64,40] (cols padded to 48)
// sW[p*128 + col*2 + j] = W2[(2p+j)*40 + col] (zero for col>=40); stride 128 (96 used + 32 pad)
__global__ void __launch_bounds__(256) gemm2_kernel(const float* __restrict__ hbn,
                                                    const float* __restrict__ W2,
                                                    float* __restrict__ h2, int n) {
  __shared__ float sW[32 * 128];  // 16 KB
  for (int i = threadIdx.x; i < 32 * 96; i += 256) {
    int p = i / 96, r = i - p * 96;
    int col = r >> 1, j = r & 1;
    sW[p * 128 + r] = (col < 40) ? W2[(2 * p + j) * 40 + col] : 0.f;
  }
  __syncthreads();
  const int wave = threadIdx.x >> 5;
  const int lane = threadIdx.x & 31;
  const int row0 = (blockIdx.x * 8 + wave) * 16;
  if (row0 >= n) return;
  const int koff = (lane >= 16) ? 2 : 0;
  const int l15 = lane & 15;
  const float* arow = hbn + (size_t)(row0 + l15) * 64 + koff;
  const float* brow = sW + (koff >> 1) * 128 + l15 * 2;
  v8f acc0 = {}, acc1 = {}, acc2 = {};
#pragma unroll 4
  for (int k = 0; k < 64; k += 4) {
    v2f a = *(const v2f*)(arow + k);
    const float* bp = brow + (k >> 1) * 128;
    v2f b0 = *(const v2f*)(bp + 0);
    v2f b1 = *(const v2f*)(bp + 32);
    v2f b2 = *(const v2f*)(bp + 64);
    acc0 = __builtin_amdgcn_wmma_f32_16x16x4_f32(false, a, false, b0, (short)0, acc0, false, false);
    acc1 = __builtin_amdgcn_wmma_f32_16x16x4_f32(false, a, false, b1, (short)0, acc1, false, false);
    acc2 = __builtin_amdgcn_wmma_f32_16x16x4_f32(false, a, false, b2, (short)0, acc2, false, false);
  }
  const int mb = (lane >> 4) * 8;
#pragma unroll
  for (int m = 0; m < 8; ++m) {
    float* orow = h2 + (size_t)(row0 + mb + m) * 40;
    orow[l15] = acc0[m];
    orow[16 + l15] = acc1[m];
    if (32 + l15 < 40) orow[32 + l15] = acc2[m];
  }
}

// ---------------- per-node attention dots + accumulator init (layer 1, H=4, C=16)
__global__ void att1_kernel(const float* __restrict__ h1,
                            const float* __restrict__ att_s, const float* __restrict__ att_d,
                            const float* __restrict__ bias1,
                            float* __restrict__ a_s, float* __restrict__ a_d,
                            float* __restrict__ amax, float* __restrict__ denom,
                            float* __restrict__ out1, int n) {
  int i = blockIdx.x * blockDim.x + threadIdx.x;
  if (i >= n) return;
  const float* hr = h1 + (size_t)i * 64;
  float* o = out1 + (size_t)i * 64;
#pragma unroll
  for (int h = 0; h < 4; ++h) {
    float ss = 0.f, dd = 0.f;
#pragma unroll
    for (int c = 0; c < 16; ++c) {
      float v = hr[h * 16 + c];
      ss += v * att_s[h * 16 + c];
      dd += v * att_d[h * 16 + c];
    }
    a_s[(size_t)i * 4 + h] = ss;
    a_d[(size_t)i * 4 + h] = dd;
    amax[(size_t)i * 4 + h] = -INFINITY;
    denom[(size_t)i * 4 + h] = 0.f;
  }
#pragma unroll
  for (int c = 0; c < 64; ++c) o[c] = bias1[c];
}

// ---------------- layer-1 edge passes (Et = E + N; e >= E means self loop)
__global__ void edge_max1(const long long* __restrict__ ei,
                          const float* __restrict__ a_s, const float* __restrict__ a_d,
                          float* __restrict__ amax, int E, int Et) {
  int e = blockIdx.x * blockDim.x + threadIdx.x;
  if (e >= Et) return;
  int s, d;
  if (e < E) { s = (int)ei[e]; d = (int)ei[(size_t)E + e]; } else { s = d = e - E; }
  const float4 as = *(const float4*)(a_s + (size_t)s * 4);
  const float4 ad = *(const float4*)(a_d + (size_t)d * 4);
  float* am = amax + (size_t)d * 4;
  atomicMaxF(am + 0, lrelu(as.x + ad.x));
  atomicMaxF(am + 1, lrelu(as.y + ad.y));
  atomicMaxF(am + 2, lrelu(as.z + ad.z));
  atomicMaxF(am + 3, lrelu(as.w + ad.w));
}

__global__ void edge_sum1(const long long* __restrict__ ei,
                          const float* __restrict__ a_s, const float* __restrict__ a_d,
                          const float* __restrict__ amax, float* __restrict__ denom,
                          int E, int Et) {
  int e = blockIdx.x * blockDim.x + threadIdx.x;
  if (e >= Et) return;
  int s, d;
  if (e < E) { s = (int)ei[e]; d = (int)ei[(size_t)E + e]; } else { s = d = e - E; }
  const float4 as = *(const float4*)(a_s + (size_t)s * 4);
  const float4 ad = *(const float4*)(a_d + (size_t)d * 4);
  const float4 am = *(const float4*)(amax + (size_t)d * 4);
  float* dn = denom + (size_t)d * 4;
  atomicAdd(dn + 0, __expf(lrelu(as.x + ad.x) - am.x));
  atomicAdd(dn + 1, __expf(lrelu(as.y + ad.y) - am.y));
  atomicAdd(dn + 2, __expf(lrelu(as.z + ad.z) - am.z));
  atomicAdd(dn + 3, __expf(lrelu(as.w + ad.w) - am.w));
}

__global__ void edge_agg1(const long long* __restrict__ ei,
                          const float* __restrict__ a_s, const float* __restrict__ a_d,
                          const float* __restrict__ amax, const float* __restrict__ denom,
                          const float* __restrict__ h1, float* __restrict__ out1,
                          int E, int Et) {
  int gid = blockIdx.x * blockDim.x + threadIdx.x;
  int e = gid >> 2, h = gid & 3;
  if (e >= Et) return;
  int s, d;
  if (e < E) { s = (int)ei[e]; d = (int)ei[(size_t)E + e]; } else { s = d = e - E; }
  float alpha = lrelu(a_s[(size_t)s * 4 + h] + a_d[(size_t)d * 4 + h]);
  float coef = __expf(alpha - amax[(size_t)d * 4 + h]) / denom[(size_t)d * 4 + h];
  const float* hs = h1 + (size_t)s * 64 + h * 16;
  float* od = out1 + (size_t)d * 64 + h * 16;
#pragma unroll
  for (int c = 0; c < 16; ++c) atomicAdd(od + c, hs[c] * coef);
}

// ---------------- BatchNorm (eval) + ReLU, in place over [N,64]
__global__ void bnrelu_kernel(float* __restrict__ out1,
                              const float* __restrict__ g, const float* __restrict__ b,
                              const float* __restrict__ mean, const float* __restrict__ var,
                              int total) {
  int i = blockIdx.x * blockDim.x + threadIdx.x;
  if (i >= total) return;
  int c = i & 63;
  float v = (out1[i] - mean[c]) * rsqrtf(var[c] + BN_EPS) * g[c] + b[c];
  out1[i] = v > 0.f ? v : 0.f;
}

// ---------------- per-node attention dots + init (layer 2, H=1, C=40)
__global__ void att2_kernel(const float* __restrict__ h2,
                            const float* __restrict__ att_s, const float* __restrict__ att_d,
                            const float* __restrict__ bias2,
                            float* __restrict__ a_s, float* __restrict__ a_d,
                            float* __restrict__ amax, float* __restrict__ denom,
                            float* __restrict__ out, int n) {
  int i = blockIdx.x * blockDim.x + threadIdx.x;
  if (i >= n) return;
  const float* hr = h2 + (size_t)i * 40;
  float* o = out + (size_t)i * 40;
  float ss = 0.f, dd = 0.f;
#pragma unroll
  for (int c = 0; c < 40; ++c) {
    float v = hr[c];
    ss += v * att_s[c];
    dd += v * att_d[c];
    o[c] = bias2[c];
  }
  a_s[i] = ss; a_d[i] = dd;
  amax[i] = -INFINITY; denom[i] = 0.f;
}

__global__ void edge_max2(const long long* __restrict__ ei,
                          const float* __restrict__ a_s, const float* __restrict__ a_d,
                          float* __restrict__ amax, int E, int Et) {
  int e = blockIdx.x * blockDim.x + threadIdx.x;
  if (e >= Et) return;
  int s, d;
  if (e < E) { s = (int)ei[e]; d = (int)ei[(size_t)E + e]; } else { s = d = e - E; }
  atomicMaxF(amax + d, lrelu(a_s[s] + a_d[d]));
}

__global__ void edge_sum2(const long long* __restrict__ ei,
                          const float* __restrict__ a_s, const float* __restrict__ a_d,
                          const float* __restrict__ amax, float* __restrict__ denom,
                          int E, int Et) {
  int e = blockIdx.x * blockDim.x + threadIdx.x;
  if (e >= Et) return;
  int s, d;
  if (e < E) { s = (int)ei[e]; d = (int)ei[(size_t)E + e]; } else { s = d = e - E; }
  atomicAdd(denom + d, __expf(lrelu(a_s[s] + a_d[d]) - amax[d]));
}

__global__ void edge_agg2(const long long* __restrict__ ei,
                          const float* __restrict__ a_s, const float* __restrict__ a_d,
                          const float* __restrict__ amax, const float* __restrict__ denom,
                          const float* __restrict__ h2, float* __restrict__ out,
                          int E, int Et) {
  int e = blockIdx.x * blockDim.x + threadIdx.x;
  if (e >= Et) return;
  int s, d;
  if (e < E) { s = (int)ei[e]; d = (int)ei[(size_t)E + e]; } else { s = d = e - E; }
  float coef = __expf(lrelu(a_s[s] + a_d[d]) - amax[d]) / denom[d];
  const float* hs = h2 + (size_t)s * 40;
  float* od = out + (size_t)d * 40;
#pragma unroll
  for (int c = 0; c < 40; ++c) atomicAdd(od + c, hs[c] * coef);
}

extern "C" void kernel_launch(void* const* d_in, const int* in_sizes, int n_in,
                              void* d_out, int out_size, void* d_ws, size_t ws_size,
                              hipStream_t stream) {
  const float* x        = (const float*)d_in[0];
  const long long* ei   = (const long long*)d_in[1];
  const float* W1       = (const float*)d_in[2];
  const float* att_src1 = (const float*)d_in[3];
  const float* att_dst1 = (const float*)d_in[4];
  const float* bias1    = (const float*)d_in[5];
  const float* bn_gamma = (const float*)d_in[6];
  const float* bn_beta  = (const float*)d_in[7];
  const float* bn_mean  = (const float*)d_in[8];
  const float* bn_var   = (const float*)d_in[9];
  const float* W2       = (const float*)d_in[10];
  const float* att_src2 = (const float*)d_in[11];
  const float* att_dst2 = (const float*)d_in[12];
  const float* bias2    = (const float*)d_in[13];
  float* out = (float*)d_out;

  const int N  = in_sizes[0] / 128;
  const int E  = in_sizes[1] / 2;
  const int Et = E + N;

  // workspace carve (floats)
  float* ws    = (float*)d_ws;
  float* h1    = ws;                         // N*64
  float* out1  = h1 + (size_t)N * 64;        // N*64
  float* a_s1  = out1 + (size_t)N * 64;      // N*4
  float* a_d1  = a_s1 + (size_t)N * 4;       // N*4
  float* amax1 = a_d1 + (size_t)N * 4;       // N*4
  float* den1  = amax1 + (size_t)N * 4;      // N*4
  float* a_s2  = den1 + (size_t)N * 4;       // N
  float* a_d2  = a_s2 + (size_t)N;           // N
  float* amax2 = a_d2 + (size_t)N;           // N
  float* den2  = amax2 + (size_t)N;          // N
  float* h2    = h1;                         // reuse h1 region after layer 1 done

  const int ntiles = (N + 15) / 16;
  const int gblk   = (ntiles + 7) / 8;
  const int nblk   = (N + 255) / 256;
  const int eblk   = (Et + 255) / 256;
  const int ehblk  = (Et * 4 + 255) / 256;
  const int bnblk  = (N * 64 + 255) / 256;

  // ---- layer 1
  gemm1_kernel<<<gblk, 256, 0, stream>>>(x, W1, h1, N);
  att1_kernel<<<nblk, 256, 0, stream>>>(h1, att_src1, att_dst1, bias1,
                                        a_s1, a_d1, amax1, den1, out1, N);
  edge_max1<<<eblk, 256, 0, stream>>>(ei, a_s1, a_d1, amax1, E, Et);
  edge_sum1<<<eblk, 256, 0, stream>>>(ei, a_s1, a_d1, amax1, den1, E, Et);
  edge_agg1<<<ehblk, 256, 0, stream>>>(ei, a_s1, a_d1, amax1, den1, h1, out1, E, Et);
  bnrelu_kernel<<<bnblk, 256, 0, stream>>>(out1, bn_gamma, bn_beta, bn_mean, bn_var, N * 64);

  // ---- layer 2
  gemm2_kernel<<<gblk, 256, 0, stream>>>(out1, W2, h2, N);
  att2_kernel<<<nblk, 256, 0, stream>>>(h2, att_src2, att_dst2, bias2,
                                        a_s2, a_d2, amax2, den2, out, N);
  edge_max2<<<eblk, 256, 0, stream>>>(ei, a_s2, a_d2, amax2, E, Et);
  edge_sum2<<<eblk, 256, 0, stream>>>(ei, a_s2, a_d2, amax2, den2, E, Et);
  edge_agg2<<<eblk, 256, 0, stream>>>(ei, a_s2, a_d2, amax2, den2, h2, out, E, Et);
}